// HarmonicLowering_28363964023074
// MI455X (gfx1250) — compile-verified
//
#include <hip/hip_runtime.h>
#include <cstdint>

// HarmonicLowering: out[b, k*C+c, f, t] = w*x[b,c,idx,t] + (1-w)*x[b,c,idx1,t]
//   idx = (f*(k+1))/4, frac = ((f*(k+1)) % 4) * 0.25, w = 1 - frac,
//   idx1 = min(idx+1, F-1)
// Shapes: x (8,32,256,512) f32 -> out (8,128,256,512) f32.
// Bandwidth-bound: CDNA5 async global->LDS staging + non-temporal b128 stores.

typedef int   v4i __attribute__((vector_size(16)));     // matches builtin param type
typedef float v4f __attribute__((ext_vector_type(4)));  // native vector for NT store

constexpr int BATCH = 8;
constexpr int C     = 32;
constexpr int F     = 256;
constexpr int T     = 512;
constexpr int KS    = 4;   // F_KERNEL_SIZE harmonics (ks = k+1)

constexpr int FT       = 16;           // output f-rows per block
constexpr int TQ       = T / 4;        // float4 chunks per t-row = 128
constexpr int NTHREADS = 256;          // 8 waves (wave32)
constexpr int MAX_ROWS = 18;           // staged input rows (worst case 17, k=4)

__global__ __launch_bounds__(NTHREADS)
void harmonic_lowering_kernel(const float* __restrict__ x, float* __restrict__ out) {
    __shared__ __align__(16) float lds[MAX_ROWS * T];   // 36 KB

    const int f0 = blockIdx.x * FT;
    const int k  = blockIdx.y / C;      // 0..3
    const int c  = blockIdx.y % C;
    const int b  = blockIdx.z;
    const int ks = k + 1;

    // Input row window [r0, r1] needed by this tile (includes clamped idx+1).
    const int r0 = (f0 * ks) >> 2;
    int r1 = ((f0 + FT - 1) * ks) >> 2;
    r1 = min(r1 + 1, F - 1);
    const int nrows = r1 - r0 + 1;      // <= 17
    const int nvec  = nrows * TQ;       // 16-byte chunks to stage

    const float* xrow = x + ((size_t)(b * C + c) * F + r0) * (size_t)T;
    const int tid = threadIdx.x;

    // ---------- Stage input window into LDS ----------
#if __has_builtin(__builtin_amdgcn_global_load_async_to_lds_b128) && \
    __has_builtin(__builtin_amdgcn_s_wait_asynccnt)
    for (int i = tid; i < nvec; i += NTHREADS) {
        const float* g = xrow + (size_t)i * 4;
        auto gp = (__attribute__((address_space(1))) v4i*)(uintptr_t)g;
        auto lp = (__attribute__((address_space(3))) v4i*)(uintptr_t)(&lds[i * 4]);
        __builtin_amdgcn_global_load_async_to_lds_b128(gp, lp, 0, 0);
    }
    __builtin_amdgcn_s_wait_asynccnt(0);
    __syncthreads();
#else
    for (int i = tid; i < nvec; i += NTHREADS) {
        ((v4f*)lds)[i] = ((const v4f*)xrow)[i];
    }
    __syncthreads();
#endif

    // Speculative prefetch of the rows the next f-tile will stage.
    if (tid == 0) {
        __builtin_prefetch(xrow + (size_t)nrows * T, 0, 0);
    }

    // ---------- Interpolate from LDS, stream NT stores ----------
    float* orow = out + ((size_t)((b * KS + k) * C + c) * F + f0) * (size_t)T;
    const v4f* ldsv = (const v4f*)lds;

    for (int i = tid; i < FT * TQ; i += NTHREADS) {
        const int fo = i >> 7;           // i / TQ
        const int q  = i & (TQ - 1);
        const int f  = f0 + fo;
        const int prod = f * ks;
        const int idx  = prod >> 2;
        const float w  = 1.0f - 0.25f * (float)(prod & 3);
        const float u  = 1.0f - w;
        const int l0 = idx - r0;
        const int l1 = min(idx + 1, F - 1) - r0;

        const v4f a = ldsv[l0 * TQ + q];
        const v4f g = ldsv[l1 * TQ + q];
        v4f o;
        o.x = w * a.x + u * g.x;
        o.y = w * a.y + u * g.y;
        o.z = w * a.z + u * g.z;
        o.w = w * a.w + u * g.w;
        __builtin_nontemporal_store(o, (v4f*)orow + i);
    }
}

extern "C" void kernel_launch(void* const* d_in, const int* in_sizes, int n_in,
                              void* d_out, int out_size, void* d_ws, size_t ws_size,
                              hipStream_t stream) {
    const float* x = (const float*)d_in[0];
    float* out     = (float*)d_out;

    dim3 grid(F / FT, KS * C, BATCH);   // (16, 128, 8)
    harmonic_lowering_kernel<<<grid, NTHREADS, 0, stream>>>(x, out);
}